// NeuralODEEncoder_24885040513701
// MI455X (gfx1250) — compile-verified
//
#include <hip/hip_runtime.h>
#include <hip/hip_bf16.h>

// NeuralODE encoder for MI455X (gfx1250).
// Only r[:,0,:] is consumed => encoder runs on 1 of 128 sequence positions.
// ODE (128 RK4 steps of a 256->512->256 tanh MLP on a 512x256 state)
// dominates: ~137 GFLOP fp32, weights L2-resident. Uses exact-fp32
// V_WMMA_F32_16X16X4_F32 (16-bit inputs would drift over 128 sequential
// steps at the reference's 1e-5 ODE tolerance).
// This revision: fragment-packed weights (coalesced b128 B-loads) + depth-2
// register software pipeline with compile-time K so loads stay >=2 chunks
// (16 WMMAs) ahead of their consuming v_wmma.

typedef __attribute__((ext_vector_type(2))) float v2f;
typedef __attribute__((ext_vector_type(4))) float v4f;
typedef __attribute__((ext_vector_type(8))) float v8f;

static __device__ __forceinline__ v8f wmma_f32_4(v2f a, v2f b, v8f c) {
  // 8 args: (neg_a, A, neg_b, B, c_mod, C, reuse_a, reuse_b)
  return __builtin_amdgcn_wmma_f32_16x16x4_f32(false, a, false, b, (short)0, c,
                                               false, false);
}

// One chunk = 8 k-values = 2 WMMA k-steps across 4 column tiles (8 wmmas).
static __device__ __forceinline__ void compute_chunk(v8f c[4], v2f alo, v2f ahi,
                                                     const v4f b[4]) {
#pragma unroll
  for (int j = 0; j < 4; ++j) {
    v2f blo = __builtin_shufflevector(b[j], b[j], 0, 1);
    c[j] = wmma_f32_4(alo, blo, c[j]);
  }
#pragma unroll
  for (int j = 0; j < 4; ++j) {
    v2f bhi = __builtin_shufflevector(b[j], b[j], 2, 3);
    c[j] = wmma_f32_4(ahi, bhi, c[j]);
  }
}

// ---- GEMM core: one wave computes a 16x64 tile of C = A(MxK) @ W ---------
// A: row-major MxK (activations).  P: weights pre-packed in per-lane WMMA
// fragment order: P[((t16*32 + lane)*(K/4) + s)*2 + u] =
//                 W[4*s + 2*(lane>>4) + u][16*t16 + (lane&15)]
// so a lane's B stream for one 16-col tile is contiguous: one b128 load
// covers 2 k-steps; adjacent lanes are K/2 floats apart.
// ISA 7.12.2 fragment layouts (32-bit, wave32):
//   A 16x4 : lane(half,idx) VGPRv = A[idx][k0 + v + 2*half]
//   B 4x16 : lane(half,idx) VGPRv = B[k0 + v + 2*half][idx]
//   C 16x16: lane(half,idx) VGPRv = C[v + 8*half][idx]
template <int K>
static __device__ __forceinline__ void gemm_core(const float* __restrict__ A,
                                                 const float* __restrict__ P,
                                                 int tm, int tn, int lane,
                                                 v8f c[4]) {
  const int idx = lane & 15;
  const int half = lane >> 4;
  const int Kh = K >> 1;  // packed floats per (tile,lane) row
  const float* ap = A + (tm * 16 + idx) * K + 2 * half;
  const float* bp[4];
#pragma unroll
  for (int j = 0; j < 4; ++j) bp[j] = P + ((tn * 4 + j) * 32 + lane) * Kh;
  c[0] = (v8f){};
  c[1] = (v8f){};
  c[2] = (v8f){};
  c[3] = (v8f){};

  // depth-2 pipeline over chunks of 8 k-values
  v2f a0lo = *(const v2f*)(ap + 0);
  v2f a0hi = *(const v2f*)(ap + 4);
  v2f a1lo = *(const v2f*)(ap + 8);
  v2f a1hi = *(const v2f*)(ap + 12);
  v4f b0[4], b1[4];
#pragma unroll
  for (int j = 0; j < 4; ++j) {
    b0[j] = *(const v4f*)(bp[j] + 0);
    b1[j] = *(const v4f*)(bp[j] + 4);
  }
#pragma unroll
  for (int k0 = 0; k0 < K - 16; k0 += 16) {
    // prefetch chunk k0+16 (consumed 2 chunks later)
    v2f na0lo = *(const v2f*)(ap + k0 + 16);
    v2f na0hi = *(const v2f*)(ap + k0 + 20);
    v4f nb0[4];
#pragma unroll
    for (int j = 0; j < 4; ++j) nb0[j] = *(const v4f*)(bp[j] + (k0 + 16) / 2);
    compute_chunk(c, a0lo, a0hi, b0);
    // prefetch chunk k0+24
    v2f na1lo = *(const v2f*)(ap + k0 + 24);
    v2f na1hi = *(const v2f*)(ap + k0 + 28);
    v4f nb1[4];
#pragma unroll
    for (int j = 0; j < 4; ++j) nb1[j] = *(const v4f*)(bp[j] + (k0 + 24) / 2);
    compute_chunk(c, a1lo, a1hi, b1);
    a0lo = na0lo; a0hi = na0hi; a1lo = na1lo; a1hi = na1hi;
#pragma unroll
    for (int j = 0; j < 4; ++j) { b0[j] = nb0[j]; b1[j] = nb1[j]; }
  }
  compute_chunk(c, a0lo, a0hi, b0);
  compute_chunk(c, a1lo, a1hi, b1);
}

// out = act(A @ W + bias).  ACT: 0=none, 1=relu, 2=tanh.
template <int ACT, int N, int K>
__global__ void wmma_gemm_act(const float* __restrict__ A,
                              const float* __restrict__ P,
                              const float* __restrict__ bias,
                              float* __restrict__ out) {
  const int lane = threadIdx.x & 31;
  const int wave = threadIdx.x >> 5;
  const int tile = blockIdx.x * (blockDim.x >> 5) + wave;
  const int tiles_n = N >> 6;
  const int tm = tile / tiles_n;
  const int tn = tile - tm * tiles_n;
  v8f c[4];
  gemm_core<K>(A, P, tm, tn, lane, c);
  const int idx = lane & 15;
  const int half = lane >> 4;
#pragma unroll
  for (int j = 0; j < 4; ++j) {
    const int col = tn * 64 + j * 16 + idx;
    const float bj = bias[col];
#pragma unroll
    for (int v = 0; v < 8; ++v) {
      float f = c[j][v] + bj;
      if (ACT == 1) f = fmaxf(f, 0.0f);
      if (ACT == 2) f = tanhf(f);
      out[(tm * 16 + v + 8 * half) * N + col] = f;
    }
  }
}

// Second GEMM of each ODE f-eval with fused RK4 stage epilogue.
// F = A @ W + bias (= k_i), then per element i:
//   stage1: acc  = F;   tin = h + (dt/2) F
//   stage2: acc += 2F;  tin = h + (dt/2) F
//   stage3: acc += 2F;  tin = h + dt F
//   stage4: h   += (dt/6)(acc + F)
template <int STAGE, int N, int K>
__global__ void wmma_gemm_rk4(const float* __restrict__ A,
                              const float* __restrict__ P,
                              const float* __restrict__ bias,
                              float* __restrict__ hbuf,
                              float* __restrict__ accbuf,
                              float* __restrict__ tinbuf, float dt) {
  const int lane = threadIdx.x & 31;
  const int wave = threadIdx.x >> 5;
  const int tile = blockIdx.x * (blockDim.x >> 5) + wave;
  const int tiles_n = N >> 6;
  const int tm = tile / tiles_n;
  const int tn = tile - tm * tiles_n;
  v8f c[4];
  gemm_core<K>(A, P, tm, tn, lane, c);
  const int idx = lane & 15;
  const int half = lane >> 4;
#pragma unroll
  for (int j = 0; j < 4; ++j) {
    const int col = tn * 64 + j * 16 + idx;
    const float bj = bias[col];
#pragma unroll
    for (int v = 0; v < 8; ++v) {
      const int i = (tm * 16 + v + 8 * half) * N + col;
      const float f = c[j][v] + bj;
      if (STAGE == 1) {
        accbuf[i] = f;
        tinbuf[i] = hbuf[i] + 0.5f * dt * f;
      } else if (STAGE == 2) {
        accbuf[i] += 2.0f * f;
        tinbuf[i] = hbuf[i] + 0.5f * dt * f;
      } else if (STAGE == 3) {
        accbuf[i] += 2.0f * f;
        tinbuf[i] = hbuf[i] + dt * f;
      } else {
        hbuf[i] = hbuf[i] + dt * (1.0f / 6.0f) * (accbuf[i] + f);
      }
    }
  }
}

// Encoder layer 1 for the single needed sequence position (n=0 after flip ->
// original index N-1).  K=2, plain VALU kernel is optimal.
__global__ void enc_layer1(const float* __restrict__ x,
                           const float* __restrict__ y,
                           const float* __restrict__ W1,
                           const float* __restrict__ b1,
                           float* __restrict__ h1, int Npts, int H) {
  const int t = blockIdx.x * blockDim.x + threadIdx.x;
  const int b = t / H;
  const int j = t - b * H;
  const float in0 = x[b * Npts + (Npts - 1)];
  const float in1 = y[b * Npts + (Npts - 1)];
  const float v = in0 * W1[j] + in1 * W1[H + j] + b1[j];
  h1[b * H + j] = fmaxf(v, 0.0f);
}

// One-time pack of W (row-major KxN) into per-lane WMMA fragment order:
//   P[tid] with tid = ((t16*32 + l)*(K/4) + s)*2 + u
//   value  = W[4*s + 2*(l>>4) + u][16*t16 + (l&15)]
__global__ void pack_weights(const float* __restrict__ W,
                             float* __restrict__ P, int K, int N) {
  const int tid = blockIdx.x * blockDim.x + threadIdx.x;
  const int kh = K >> 1;
  const int u = tid & 1;
  const int s = (tid >> 1) % (K >> 2);
  const int l = (tid / kh) % 32;
  const int t = tid / (kh * 32);
  const int k = 4 * s + 2 * (l >> 4) + u;
  const int n = 16 * t + (l & 15);
  P[tid] = W[k * N + n];
}

extern "C" void kernel_launch(void* const* d_in, const int* in_sizes, int n_in,
                              void* d_out, int out_size, void* d_ws,
                              size_t ws_size, hipStream_t stream) {
  (void)in_sizes; (void)n_in; (void)out_size; (void)ws_size;
  const float* x = (const float*)d_in[0];       // (512,128,1)
  const float* y = (const float*)d_in[1];       // (512,128,1)
  const float* enc_W1 = (const float*)d_in[2];  // (2,512)
  const float* enc_b1 = (const float*)d_in[3];
  const float* enc_W2 = (const float*)d_in[4];  // (512,512)
  const float* enc_b2 = (const float*)d_in[5];
  const float* enc_W3 = (const float*)d_in[6];  // (512,256)
  const float* enc_b3 = (const float*)d_in[7];
  const float* ode_W1 = (const float*)d_in[8];  // (256,512)
  const float* ode_b1 = (const float*)d_in[9];
  const float* ode_W2 = (const float*)d_in[10]; // (512,256)
  const float* ode_b2 = (const float*)d_in[11];

  const int Bb = 512, Npts = 128, Hh = 512, Rr = 256;
  float* ws = (float*)d_ws;
  float* h = ws;                 // B*R state
  float* acc = h + Bb * Rr;      // B*R RK4 accumulator
  float* tin = acc + Bb * Rr;    // B*R stage input
  float* z = tin + Bb * Rr;      // B*H hidden (also reused as enc h2)
  float* h1 = z + Bb * Hh;       // B*H encoder hidden 1
  float* PE2 = h1 + Bb * Hh;     // packed enc_W2 (512x512)
  float* PE3 = PE2 + Hh * Hh;    // packed enc_W3 (512x256)
  float* P1 = PE3 + Hh * Rr;     // packed ode_W1 (256x512)
  float* P2 = P1 + Rr * Hh;      // packed ode_W2 (512x256)

  const int TB = 128;  // 4 waves/block -> spread small GEMMs over more WGPs

  // ---- one-time weight fragment packing (~2.5 MB total) ----
  pack_weights<<<(Hh * Hh) / 256, 256, 0, stream>>>(enc_W2, PE2, Hh, Hh);
  pack_weights<<<(Hh * Rr) / 256, 256, 0, stream>>>(enc_W3, PE3, Hh, Rr);
  pack_weights<<<(Rr * Hh) / 256, 256, 0, stream>>>(ode_W1, P1, Rr, Hh);
  pack_weights<<<(Hh * Rr) / 256, 256, 0, stream>>>(ode_W2, P2, Hh, Rr);

  // ---- encoder on the single required row per batch element ----
  enc_layer1<<<(Bb * Hh) / 256, 256, 0, stream>>>(x, y, enc_W1, enc_b1, h1,
                                                  Npts, Hh);
  // tiles = (M/16)*(N/64); blocks = tiles / (TB/32)
  // h2 = relu(h1 @ W2 + b2): 512x512x512 -> 256 tiles -> 64 blocks
  wmma_gemm_act<1, 512, 512><<<64, TB, 0, stream>>>(h1, PE2, enc_b2, z);
  // h0 = h2 @ W3 + b3: 512x256x512 -> 128 tiles -> 32 blocks
  wmma_gemm_act<0, 256, 512><<<32, TB, 0, stream>>>(z, PE3, enc_b3, h);

  // ---- RK4 integration of dh/dt = tanh(h@W1+b1)@W2+b2 over t in [0,1] ----
  const int NSTEPS = 128;
  const float dt = 1.0f / (float)NSTEPS;
  for (int s = 0; s < NSTEPS; ++s) {
    wmma_gemm_act<2, 512, 256><<<64, TB, 0, stream>>>(h, P1, ode_b1, z);
    wmma_gemm_rk4<1, 256, 512><<<32, TB, 0, stream>>>(z, P2, ode_b2, h, acc,
                                                      tin, dt);
    wmma_gemm_act<2, 512, 256><<<64, TB, 0, stream>>>(tin, P1, ode_b1, z);
    wmma_gemm_rk4<2, 256, 512><<<32, TB, 0, stream>>>(z, P2, ode_b2, h, acc,
                                                      tin, dt);
    wmma_gemm_act<2, 512, 256><<<64, TB, 0, stream>>>(tin, P1, ode_b1, z);
    wmma_gemm_rk4<3, 256, 512><<<32, TB, 0, stream>>>(z, P2, ode_b2, h, acc,
                                                      tin, dt);
    wmma_gemm_act<2, 512, 256><<<64, TB, 0, stream>>>(tin, P1, ode_b1, z);
    wmma_gemm_rk4<4, 256, 512><<<32, TB, 0, stream>>>(z, P2, ode_b2, h, acc,
                                                      tin, dt);
  }

  hipMemcpyAsync(d_out, h, (size_t)Bb * Rr * sizeof(float),
                 hipMemcpyDeviceToDevice, stream);
}